// Gfusedmax_65051574665523
// MI455X (gfx1250) — compile-verified
//
#include <hip/hip_runtime.h>
#include <hip/hip_bf16.h>
#include <cstdint>

// Problem constants (from reference)
#define B_DIM 4096
#define M_DIM 128
#define LDH   132          // padded LDS row stride (floats): 132*4B = 16B-aligned rows
#define NS_ITERS 14        // Newton-Schulz iterations (quadratic conv., diag-dominant H)
#define ADMM_ITERS 50
#define RHO_C 1.0f
#define LAM_C 1.0f

typedef __attribute__((ext_vector_type(2))) float v2f;
typedef __attribute__((ext_vector_type(8))) float v8f;
typedef __attribute__((ext_vector_type(4))) unsigned int v4u;
typedef __attribute__((ext_vector_type(4))) int v4i;
typedef __attribute__((ext_vector_type(8))) int v8i;

#if defined(__gfx1250__) && __has_builtin(__builtin_amdgcn_tensor_load_to_lds)
#define HAVE_TDM 1
#else
#define HAVE_TDM 0
#endif

// 128x128x128 f32 GEMM in LDS using V_WMMA_F32_16X16X4_F32.
// C = A*B            (Zb == nullptr)
// C = 2*Z - A*B      (Zb != nullptr)   [Newton-Schulz combine]
// 8 waves: wave w owns output tile-row w (16 rows), loops 8 tiles along N.
// A fragments for the whole tile-row are cached in VGPRs; BOTH k-loops are
// fully unrolled so afr[] has only constant indices -> stays in registers
// (a partially-unrolled loop previously forced it into scratch).
// Must be called by all 256 threads with uniform args (EXEC all-ones for WMMA).
__device__ __forceinline__ void gemm128(float* __restrict__ Cb,
                                        const float* __restrict__ Ab,
                                        const float* __restrict__ Bb,
                                        const float* __restrict__ Zb,
                                        int tid) {
  const int wave = tid >> 5;      // 0..7 -> output tile row
  const int lane = tid & 31;
  const int l16  = lane & 15;
  const int hi   = lane >> 4;     // 0/1 : K sub-pair selector per ISA A-frag layout
  const int mrow = wave * 16 + l16;
  const int kofs = hi * 2;

  // A 16x4 frags: lanes0-15 hold K={0,1}, lanes16-31 hold K={2,3}
  v2f afr[32];
  #pragma unroll
  for (int kk = 0; kk < 32; ++kk) {
    afr[kk].x = Ab[mrow * LDH + kk * 4 + kofs];
    afr[kk].y = Ab[mrow * LDH + kk * 4 + kofs + 1];
  }

  #pragma unroll 1
  for (int nT = 0; nT < 8; ++nT) {
    const int ncol = nT * 16 + l16;
    v8f acc = {};
    #pragma unroll   // FULL unroll: constant afr[] indices -> register promotion
    for (int kk = 0; kk < 32; ++kk) {
      v2f bf;  // B 4x16 frag: row K striped across lanes within each VGPR
      bf.x = Bb[(kk * 4 + kofs) * LDH + ncol];
      bf.y = Bb[(kk * 4 + kofs + 1) * LDH + ncol];
      acc = __builtin_amdgcn_wmma_f32_16x16x4_f32(
          /*neg_a=*/false, afr[kk], /*neg_b=*/false, bf,
          /*c_mod=*/(short)0, acc, /*reuse_a=*/false, /*reuse_b=*/false);
    }
    // C/D layout: VGPR g -> M = g (lanes0-15) or g+8 (lanes16-31), N = lane&15
    #pragma unroll
    for (int g = 0; g < 8; ++g) {
      const int row = wave * 16 + hi * 8 + g;
      const int idx = row * LDH + ncol;
      const float v = acc[g];
      Cb[idx] = Zb ? (2.0f * Zb[idx] - v) : v;
    }
  }
}

__global__ __launch_bounds__(256, 1)
void Gfusedmax_65051574665523_kernel(const float* __restrict__ x,
                                     const float* __restrict__ A,
                                     float* __restrict__ out) {
  extern __shared__ char smem_raw[];
  float* bufH = (float*)smem_raw;            // H
  float* bufX = bufH + M_DIM * LDH;          // X ping
  float* bufT = bufX + M_DIM * LDH;          // T = H*X   (later w)
  float* bufU = bufT + M_DIM * LDH;          // raw-A staging, then X pong
  float* sx   = bufU + M_DIM * LDH;          // [128] x row
  float* su   = sx + M_DIM;                  // [128] u
  float* srhs = su + M_DIM;                  // [128] rhs
  float* sred = srhs + M_DIM;                // [256] reductions / sorted z
  float* sdeg = sred + 256;                  // [128] degree, later tau broadcast
  uint32_t* sMask = (uint32_t*)(sdeg + M_DIM); // [128][4] adjacency bitmask

  const int tid = threadIdx.x;
  const int b   = blockIdx.x;
  const float* xg = x + (size_t)b * M_DIM;
  const float* Ag = A + (size_t)b * M_DIM * M_DIM;

  if (tid < M_DIM) sx[tid] = xg[tid];

#if HAVE_TDM
  // ---- Phase 0: TDM DMA of A[b] (128x128 f32) into LDS with row padding ----
  // pad_interval=6 (every 128 dwords) + pad_amount=3 (+4 dwords) => stride 132,
  // exactly LDH, so LDS reads below are bank-conflict-free.
  if (tid < 32) {
    const uint64_t ga = (uint64_t)(uintptr_t)Ag;
    const unsigned lds_addr =
        __builtin_amdgcn_groupstaticsize() +
        (unsigned)(3u * M_DIM * LDH * sizeof(float));   // bufU offset
    v4u g0;
    g0.x = 1u;                                   // count=1: valid descriptor
    g0.y = lds_addr;                             // lds_addr [63:32]
    g0.z = (unsigned)(ga & 0xffffffffu);         // global_addr lo
    g0.w = (unsigned)(ga >> 32) | (2u << 30);    // global_addr hi | type=2
    v8i g1;
    g1[0] = (int)((2u << 16)      // data_size = 4B
                  | (1u << 20)    // pad_enable
                  | (6u << 22)    // pad_interval: 128 dwords
                  | (3u << 25));  // pad_amount: 4 dwords
    g1[1] = (int)(128u << 16);    // tensor_dim0[15:0] in bits[63:48]
    g1[2] = (int)(128u << 16);    // tensor_dim1[15:0] in bits[95:80]
    g1[3] = (int)(128u << 16);    // tile_dim0 in bits[127:112]
    g1[4] = 128;                  // tile_dim1 in bits[143:128]
    g1[5] = 128;                  // tensor_dim0_stride[31:0]
    g1[6] = 0;
    g1[7] = 0;
    v4i z4 = {0, 0, 0, 0};
#if __clang_major__ >= 23
    v8i z8 = {0, 0, 0, 0, 0, 0, 0, 0};
    __builtin_amdgcn_tensor_load_to_lds(g0, g1, z4, z4, z8, 0);
#else
    __builtin_amdgcn_tensor_load_to_lds(g0, g1, z4, z4, 0);
#endif
    __builtin_amdgcn_s_wait_tensorcnt(0);
  }
  __syncthreads();
  const float* Asrc = bufU;
  const int ASTRIDE = LDH;
#else
  const float* Asrc = Ag;
  const int ASTRIDE = M_DIM;
#endif

  // ---- Phase 1: build mask bits, degree, H, X0 = D^-1 ----
  const int i2 = tid >> 1;   // row (2 threads per row)
  const int h2 = tid & 1;    // column half
  {
    float rs = 0.f;
    uint32_t w0 = 0u, w1 = 0u;
    const float* Arow = Asrc + i2 * ASTRIDE + h2 * 64;
    for (int j = 0; j < 64; j += 4) {
      float4 a4 = *(const float4*)(Arow + j);
      uint32_t b0 = a4.x > 0.f ? 1u : 0u;
      uint32_t b1 = a4.y > 0.f ? 1u : 0u;
      uint32_t b2 = a4.z > 0.f ? 1u : 0u;
      uint32_t b3 = a4.w > 0.f ? 1u : 0u;
      rs += (float)(b0 + b1 + b2 + b3);
      const uint32_t nib = b0 | (b1 << 1) | (b2 << 2) | (b3 << 3);
      if (j < 32) w0 |= nib << j; else w1 |= nib << (j - 32);
    }
    sMask[i2 * 4 + h2 * 2 + 0] = w0;
    sMask[i2 * 4 + h2 * 2 + 1] = w1;
    sred[tid] = rs;
  }
  __syncthreads();
  if (h2 == 0) sdeg[i2] = 2.0f * (sred[tid] + sred[tid + 1]);  // deg = row+col sum
  __syncthreads();
  {
    const float dii = 1.0f + RHO_C * sdeg[i2];
    const uint32_t mw0 = sMask[i2 * 4 + h2 * 2 + 0];
    const uint32_t mw1 = sMask[i2 * 4 + h2 * 2 + 1];
    for (int jj = 0; jj < 64; ++jj) {
      const int j = h2 * 64 + jj;
      const uint32_t bit = ((jj < 32 ? (mw0 >> jj) : (mw1 >> (jj - 32))) & 1u);
      bufH[i2 * LDH + j] = (j == i2) ? dii : (bit ? -2.0f * RHO_C : 0.0f);
      bufX[i2 * LDH + j] = (j == i2) ? (1.0f / dii) : 0.0f;
    }
  }
  __syncthreads();

  // ---- Phase 2: Newton-Schulz inversion, X <- X(2I - H X), via f32 WMMA ----
  float* Xc = bufX;
  float* Xn = bufU;   // raw A no longer needed when this is first written
  for (int it = 0; it < NS_ITERS; ++it) {
    gemm128(bufT, bufH, Xc, nullptr, tid);   // T = H * X
    __syncthreads();
    gemm128(Xn, Xc, bufT, Xc, tid);          // Xn = 2*X - X*T
    __syncthreads();
    float* t = Xc; Xc = Xn; Xn = t;
  }
  const float* Hinv = Xc;   // converged inverse
  float* dM = Xn;           // recycle pong buffer for d
  float* wM = bufT;         // recycle T buffer for w

  // ---- Phase 3: ADMM init  d0 = m*(x_i - x_j), w0 = 0 ----
  for (int e = tid; e < M_DIM * M_DIM; e += 256) {
    const int ii = e >> 7, jj = e & 127;
    const uint32_t bit = (sMask[ii * 4 + (jj >> 5)] >> (jj & 31)) & 1u;
    dM[ii * LDH + jj] = bit ? (sx[ii] - sx[jj]) : 0.0f;
    wM[ii * LDH + jj] = 0.0f;
  }
  __syncthreads();

  // ---- Phase 4: 50 ADMM iterations ----
  for (int it = 0; it < ADMM_ITERS; ++it) {
    // rhs = x + 2*rho*rowsum(d - w)   (antisymmetry: sum(-1)-sum(-2) = 2*rowsum)
    float p = 0.f;
    for (int jj = 0; jj < 64; ++jj) {
      const int j = h2 * 64 + jj;
      p += dM[i2 * LDH + j] - wM[i2 * LDH + j];
    }
    sred[tid] = p;
    __syncthreads();
    if (h2 == 0) srhs[i2] = sx[i2] + 2.0f * RHO_C * (sred[tid] + sred[tid + 1]);
    __syncthreads();
    // u = Hinv @ rhs (GEMV, 2 threads/row)
    float acc = 0.f;
    for (int jj = 0; jj < 64; ++jj) {
      const int j = h2 * 64 + jj;
      acc += Hinv[i2 * LDH + j] * srhs[j];
    }
    sred[tid] = acc;
    __syncthreads();
    if (h2 == 0) su[i2] = sred[tid] + sred[tid + 1];
    __syncthreads();
    // d = m*shrink(Du + w, lam/rho);  w = m*(w + Du - d)
    for (int e = tid; e < M_DIM * M_DIM; e += 256) {
      const int ii = e >> 7, jj = e & 127;
      const uint32_t bit = (sMask[ii * 4 + (jj >> 5)] >> (jj & 31)) & 1u;
      const float Du = su[ii] - su[jj];
      const float w  = wM[ii * LDH + jj];
      const float v  = Du + w;
      const float mag = fabsf(v) - (LAM_C / RHO_C);
      const float dnew = (bit && mag > 0.f) ? (v > 0.f ? mag : -mag) : 0.f;
      const float wnew = bit ? (w + Du - dnew) : 0.f;
      dM[ii * LDH + jj] = dnew;
      wM[ii * LDH + jj] = wnew;
    }
    __syncthreads();
  }

  // ---- Phase 5: sparsemax over u (gamma = 1) ----
  if (tid < M_DIM) {
    const float zi = su[tid];
    int pos = 0;
    for (int j = 0; j < M_DIM; ++j) {
      const float zj = su[j];
      pos += (zj > zi) || (zj == zi && j < tid);   // stable descending rank
    }
    sred[pos] = zi;
  }
  __syncthreads();
  if (tid == 0) {
    float cs = 0.f, ssum = 0.f, kc = 0.f;
    for (int j = 0; j < M_DIM; ++j) {
      const float zs = sred[j];
      cs += zs;
      if (1.0f + (float)(j + 1) * zs > cs) { ssum += zs; kc += 1.0f; }
    }
    sdeg[0] = (ssum - 1.0f) / kc;   // tau
  }
  __syncthreads();
  if (tid < M_DIM) {
    const float v = su[tid] - sdeg[0];
    out[(size_t)b * M_DIM + tid] = v > 0.f ? v : 0.f;
  }
}

extern "C" void kernel_launch(void* const* d_in, const int* in_sizes, int n_in,
                              void* d_out, int out_size, void* d_ws, size_t ws_size,
                              hipStream_t stream) {
  (void)in_sizes; (void)n_in; (void)out_size; (void)d_ws; (void)ws_size;
  const float* x = (const float*)d_in[0];
  const float* A = (const float*)d_in[1];
  float* out = (float*)d_out;

  constexpr size_t SMEM =
      (size_t)(4 * M_DIM * LDH            // H, X, T, U
               + 3 * M_DIM                // sx, su, srhs
               + 256                      // sred
               + M_DIM) * sizeof(float)   // sdeg
      + (size_t)M_DIM * 4 * sizeof(uint32_t);  // mask bits

  hipFuncSetAttribute(reinterpret_cast<const void*>(Gfusedmax_65051574665523_kernel),
                      hipFuncAttributeMaxDynamicSharedMemorySize, (int)SMEM);
  Gfusedmax_65051574665523_kernel<<<B_DIM, 256, SMEM, stream>>>(x, A, out);
}